// _OptNet_for_LLLLGG_9414568313222
// MI455X (gfx1250) — compile-verified
//
#include <hip/hip_runtime.h>
#include <stdint.h>
#include <stddef.h>

#ifndef __has_builtin
#define __has_builtin(x) 0
#endif

#if __has_builtin(__builtin_amdgcn_global_load_async_to_lds_b32)
#define HAVE_ASYNC_LDS 1
#else
#define HAVE_ASYNC_LDS 0
#endif

typedef __attribute__((address_space(1))) int g_i32;  // global i32
typedef __attribute__((address_space(3))) int l_i32;  // LDS i32

static constexpr int   NP    = 6;            // plan dimension n
static constexpr int   NC    = 62;           // core constraints
static constexpr int   M     = NC + 2 * NP;  // 74 total constraints
static constexpr int   EA    = NC * NP;      // 372 elements of A per batch
static constexpr int   NITER = 20;
static constexpr float SIGMA = 0.1f;
static constexpr float BIGV  = 1e9f;

// ---------------------------------------------------------------------------
// Kernel 1a: tiled transpose with scale: out[c][r] = scale * in[r][c]
// in: [rows][cols], out: [cols][rows]. Stages the tile through LDS using the
// CDNA5 async global->LDS path when available.
// ---------------------------------------------------------------------------
__global__ __launch_bounds__(256) void k_transpose_scale(
    const float* __restrict__ in, float* __restrict__ out,
    int rows, int cols, float scale)
{
  __shared__ float tile[32][33];
  const int c0 = blockIdx.x * 32, r0 = blockIdx.y * 32;
  const int tx = threadIdx.x, ty = threadIdx.y;  // blockDim = (32, 8)

#pragma unroll
  for (int i = 0; i < 32; i += 8) {
    int r = r0 + ty + i, c = c0 + tx;
    if (r < rows && c < cols) {
#if HAVE_ASYNC_LDS
      __builtin_amdgcn_global_load_async_to_lds_b32(
          (g_i32*)(const void*)(in + (size_t)r * cols + c),
          (l_i32*)&tile[ty + i][tx], 0, 0);
#else
      tile[ty + i][tx] = in[(size_t)r * cols + c];
#endif
    }
  }
#if HAVE_ASYNC_LDS
#if __has_builtin(__builtin_amdgcn_s_wait_asynccnt)
  __builtin_amdgcn_s_wait_asynccnt(0);
#else
  asm volatile("s_wait_asynccnt 0" ::: "memory");
#endif
#endif
  __syncthreads();

#pragma unroll
  for (int i = 0; i < 32; i += 8) {
    int c = c0 + ty + i, r = r0 + tx;
    if (c < cols && r < rows)
      out[(size_t)c * rows + r] = scale * tile[tx][ty + i];
  }
}

// ---------------------------------------------------------------------------
// Kernel 1b: pack h rows 62..73 (= b then 0) and q = -2*payment_vcg, both
// element-major [elem][B].
// ---------------------------------------------------------------------------
__global__ __launch_bounds__(256) void k_pack_small(
    const float* __restrict__ bnd, const float* __restrict__ pv,
    float* __restrict__ Ht, float* __restrict__ Qt, int nB)
{
  int bb = blockIdx.x * 256 + threadIdx.x;
  if (bb >= nB) return;
#pragma unroll
  for (int i = 0; i < NP; ++i) {
    Ht[(size_t)(NC + i) * nB + bb]      = bnd[(size_t)bb * NP + i];
    Ht[(size_t)(NC + NP + i) * nB + bb] = 0.0f;
    Qt[(size_t)i * nB + bb]             = -2.0f * pv[(size_t)bb * NP + i];
  }
}

// ---------------------------------------------------------------------------
// Kernel 2: batched primal-dual IPM, one batch per lane (wave32).
// PACKED=true : G0 = Gt[EA][nB] (= -A transposed), H0 = Ht[M][nB], Q0 = Qt[NP][nB]
// PACKED=false: G0 = A[nB][EA], H0 = beta[nB][NC], Bv = b[nB][NP], Q0 = pv[nB][NP]
// Per-lane state vectors s, lam, r_prim live in LDS (stride WG -> conflict-free).
// ---------------------------------------------------------------------------
template <int WG, bool PACKED>
__global__ __launch_bounds__(WG) void k_ipm(
    const float* __restrict__ G0, const float* __restrict__ H0,
    const float* __restrict__ Bv, const float* __restrict__ Q0,
    float* __restrict__ out, int nB)
{
  __shared__ float sh[3 * M * WG];
  const int tid = threadIdx.x;
  const int bb  = blockIdx.x * WG + tid;
  if (bb >= nB) return;
  float* Ssh = sh;              // slack s
  float* Lsh = sh + M * WG;     // lambda
  float* Rsh = sh + 2 * M * WG; // r_prim, then reused for ds

  float z[NP], q[NP];
#pragma unroll
  for (int i = 0; i < NP; ++i) {
    z[i] = 0.0f;
    q[i] = PACKED ? Q0[(size_t)i * nB + bb] : -2.0f * Q0[(size_t)bb * NP + i];
  }
  for (int e = 0; e < M; ++e) { Ssh[e * WG + tid] = 1.0f; Lsh[e * WG + tid] = 1.0f; }

  for (int it = 0; it < NITER; ++it) {
    // ---- mu = mean(s*lam)
    float mu = 0.0f;
    for (int e = 0; e < M; ++e) mu += Ssh[e * WG + tid] * Lsh[e * WG + tid];
    mu *= (1.0f / (float)M);
    const float sigmu = SIGMA * mu;

    float acc[NP], Kf[NP][NP];
#pragma unroll
    for (int i = 0; i < NP; ++i) {
      acc[i] = 0.0f;
#pragma unroll
      for (int l = 0; l < NP; ++l) Kf[i][l] = 0.0f;
    }

    // ---- pass 1 over core rows: K += w g g', acc += g*(w*rp + sigmu/s)
    for (int j = 0; j < NC; ++j) {
      float g[NP];
      if (PACKED) {
        if (j + 4 < NC)
          __builtin_prefetch(&G0[(size_t)((j + 4) * NP) * nB + bb], 0, 0);
#pragma unroll
        for (int k = 0; k < NP; ++k) g[k] = G0[(size_t)(j * NP + k) * nB + bb];
      } else {
#pragma unroll
        for (int k = 0; k < NP; ++k) g[k] = -G0[(size_t)bb * EA + j * NP + k];
      }
      float sj = Ssh[j * WG + tid], lj = Lsh[j * WG + tid];
      float hj = PACKED ? H0[(size_t)j * nB + bb] : -H0[(size_t)bb * NC + j];
      float invs = 1.0f / sj;
      float w = lj * invs;
      float gz = 0.0f;
#pragma unroll
      for (int k = 0; k < NP; ++k) gz += g[k] * z[k];
      float rp = gz + sj - hj;
      float cj = w * rp + sigmu * invs;  // lam + (w*rp - r_cent/s) simplified
#pragma unroll
      for (int i = 0; i < NP; ++i) acc[i] += g[i] * cj;
#pragma unroll
      for (int i = 0; i < NP; ++i) {
        float wgi = w * g[i];
#pragma unroll
        for (int l = i; l < NP; ++l) Kf[i][l] += wgi * g[l];
      }
      Rsh[j * WG + tid] = rp;
    }
    // ---- pass 1 over identity rows: g = +e_i (h=b), g = -e_i (h=0)
#pragma unroll
    for (int i = 0; i < NP; ++i) {
#pragma unroll
      for (int t2 = 0; t2 < 2; ++t2) {
        int j = NC + t2 * NP + i;
        float sj = Ssh[j * WG + tid], lj = Lsh[j * WG + tid];
        float hj = PACKED ? H0[(size_t)j * nB + bb]
                          : (t2 == 0 ? Bv[(size_t)bb * NP + i] : 0.0f);
        float invs = 1.0f / sj, w = lj * invs;
        float rp = (t2 == 0 ? z[i] : -z[i]) + sj - hj;
        float cj = w * rp + sigmu * invs;
        if (t2 == 0) acc[i] += cj; else acc[i] -= cj;
        Kf[i][i] += w;
        Rsh[j * WG + tid] = rp;
      }
    }

    float rhs[NP];
#pragma unroll
    for (int i = 0; i < NP; ++i) {
      rhs[i] = -(2.0f * z[i] + q[i] + acc[i]);
      Kf[i][i] += 2.0f;  // Q = 2I
    }
#pragma unroll
    for (int i = 0; i < NP; ++i)
#pragma unroll
      for (int l = i + 1; l < NP; ++l) Kf[l][i] = Kf[i][l];

    // ---- Cholesky K = L L^T (in place, lower), then two triangular solves
#pragma unroll
    for (int c = 0; c < NP; ++c) {
      float d = Kf[c][c];
      for (int p = 0; p < c; ++p) d -= Kf[c][p] * Kf[c][p];
      d = sqrtf(fmaxf(d, 1e-30f));
      Kf[c][c] = d;
      float inv = 1.0f / d;
      for (int r = c + 1; r < NP; ++r) {
        float v = Kf[r][c];
        for (int p = 0; p < c; ++p) v -= Kf[r][p] * Kf[c][p];
        Kf[r][c] = v * inv;
      }
    }
    float dz[NP];
#pragma unroll
    for (int r = 0; r < NP; ++r) {       // L y = rhs (y stored into rhs)
      float v = rhs[r];
      for (int p = 0; p < r; ++p) v -= Kf[r][p] * rhs[p];
      rhs[r] = v / Kf[r][r];
    }
#pragma unroll
    for (int r = NP - 1; r >= 0; --r) {  // L^T dz = y
      float v = rhs[r];
      for (int p = r + 1; p < NP; ++p) v -= Kf[p][r] * dz[p];
      dz[r] = v / Kf[r][r];
    }

    // ---- pass 2: ds, dlam, fraction-to-boundary ratio
    float rmin = BIGV;
    for (int j = 0; j < NC; ++j) {
      float g[NP];
      if (PACKED) {
#pragma unroll
        for (int k = 0; k < NP; ++k) g[k] = G0[(size_t)(j * NP + k) * nB + bb];
      } else {
#pragma unroll
        for (int k = 0; k < NP; ++k) g[k] = -G0[(size_t)bb * EA + j * NP + k];
      }
      float gdz = 0.0f;
#pragma unroll
      for (int k = 0; k < NP; ++k) gdz += g[k] * dz[k];
      float rp  = Rsh[j * WG + tid];
      float dsj = -rp - gdz;
      float sj = Ssh[j * WG + tid], lj = Lsh[j * WG + tid];
      float rc  = sj * lj - sigmu;
      float dlj = -(rc + lj * dsj) / sj;
      if (dsj < 0.0f) rmin = fminf(rmin, -sj / dsj);
      if (dlj < 0.0f) rmin = fminf(rmin, -lj / dlj);
      Rsh[j * WG + tid] = dsj;  // overwrite rp with ds
    }
#pragma unroll
    for (int i = 0; i < NP; ++i) {
#pragma unroll
      for (int t2 = 0; t2 < 2; ++t2) {
        int j = NC + t2 * NP + i;
        float gdz = (t2 == 0) ? dz[i] : -dz[i];
        float rp  = Rsh[j * WG + tid];
        float dsj = -rp - gdz;
        float sj = Ssh[j * WG + tid], lj = Lsh[j * WG + tid];
        float rc  = sj * lj - sigmu;
        float dlj = -(rc + lj * dsj) / sj;
        if (dsj < 0.0f) rmin = fminf(rmin, -sj / dsj);
        if (dlj < 0.0f) rmin = fminf(rmin, -lj / dlj);
        Rsh[j * WG + tid] = dsj;
      }
    }
    const float alpha = fminf(1.0f, 0.99f * rmin);

    // ---- pass 3: state update (dlam recomputed from stored ds)
    for (int j = 0; j < M; ++j) {
      float sj = Ssh[j * WG + tid], lj = Lsh[j * WG + tid];
      float dsj = Rsh[j * WG + tid];
      float rc  = sj * lj - sigmu;
      float dlj = -(rc + lj * dsj) / sj;
      Ssh[j * WG + tid] = sj + alpha * dsj;
      Lsh[j * WG + tid] = lj + alpha * dlj;
    }
#pragma unroll
    for (int i = 0; i < NP; ++i) z[i] += alpha * dz[i];
  }

#pragma unroll
  for (int i = 0; i < NP; ++i) out[(size_t)bb * NP + i] = z[i];
}

// ---------------------------------------------------------------------------
extern "C" void kernel_launch(void* const* d_in, const int* in_sizes, int n_in,
                              void* d_out, int out_size, void* d_ws, size_t ws_size,
                              hipStream_t stream) {
  const float* A    = (const float*)d_in[0];  // [B, 62, 6]
  const float* beta = (const float*)d_in[1];  // [B, 62]
  const float* bnd  = (const float*)d_in[2];  // [B, 6]
  const float* pv   = (const float*)d_in[3];  // [B, 6]
  float* out = (float*)d_out;

  const int nB = in_sizes[0] / EA;
  const size_t need = (size_t)(EA + M + NP) * (size_t)nB * sizeof(float);

  constexpr int WG = 64;  // 2 waves; 56.8 KB LDS -> several workgroups per WGP
  const int nblk = (nB + WG - 1) / WG;

  if (ws_size >= need) {
    float* Gt = (float*)d_ws;                 // [372][nB], = -A^T (G core rows)
    float* Ht = Gt + (size_t)EA * nB;         // [74][nB]
    float* Qt = Ht + (size_t)M * nB;          // [6][nB]
    dim3 blk(32, 8);
    k_transpose_scale<<<dim3((EA + 31) / 32, (nB + 31) / 32), blk, 0, stream>>>(
        A, Gt, nB, EA, -1.0f);
    k_transpose_scale<<<dim3((NC + 31) / 32, (nB + 31) / 32), blk, 0, stream>>>(
        beta, Ht, nB, NC, -1.0f);
    k_pack_small<<<(nB + 255) / 256, 256, 0, stream>>>(bnd, pv, Ht, Qt, nB);
    k_ipm<WG, true><<<nblk, WG, 0, stream>>>(Gt, Ht, nullptr, Qt, out, nB);
  } else {
    k_ipm<WG, false><<<nblk, WG, 0, stream>>>(A, beta, bnd, pv, out, nB);
  }
}